// FusedGCN_21543555956848
// MI455X (gfx1250) — compile-verified
//
#include <hip/hip_runtime.h>

typedef __attribute__((ext_vector_type(2))) float v2f;
typedef __attribute__((ext_vector_type(8))) float v8f;

#define GCN_N_NODES 100000
#define GCN_N_EDGES 3200000
#define GCN_FEAT    256
#define GCN_EMB     128
#define GCN_NCLS    64

// -------------------------------------------------------------------------
// Build CSR row_ptr from sorted COO rows (deterministic, atomic-free).
// -------------------------------------------------------------------------
__global__ void gcn_build_row_ptr(const int* __restrict__ rows,
                                  int* __restrict__ row_ptr,
                                  int n_edges, int n_nodes) {
    int e = blockIdx.x * blockDim.x + threadIdx.x;
    if (e >= n_edges) return;
    int r    = rows[e];
    int prev = (e == 0) ? -1 : rows[e - 1];
    for (int rr = prev + 1; rr <= r; ++rr) row_ptr[rr] = e;
    if (e == n_edges - 1) {
        for (int rr = r + 1; rr <= n_nodes; ++rr) row_ptr[rr] = n_edges;
    }
}

// -------------------------------------------------------------------------
// Dense GEMM C[M,N] = A[M,K] * B[K,N], row-major f32, exact f32 math via
// V_WMMA_F32_16X16X4_F32. One wave computes one 16x16 C tile; K in steps
// of 4. VGPR layouts per CDNA5 ISA 7.12.2:
//   A 16x4 : lane l holds row (l&15); VGPR0 -> K = 2*(l>=16), VGPR1 -> K+1
//   B 4x16 : lane l holds col (l&15); VGPR0 -> K = 2*(l>=16), VGPR1 -> K+1
//   C 16x16: VGPR v -> row v + 8*(l>=16), col (l&15)
// Tile guard is wave-uniform, so EXEC stays all-1s inside WMMA (required).
// -------------------------------------------------------------------------
__global__ void gcn_gemm_wmma_f32(const float* __restrict__ A,
                                  const float* __restrict__ B,
                                  float* __restrict__ C,
                                  int M, int N, int K) {
    const int lane = threadIdx.x & 31;
    const int wave = threadIdx.x >> 5;
    const int hi   = lane >> 4;       // 0 for lanes 0-15, 1 for lanes 16-31
    const int lm   = lane & 15;

    const int Mtiles = M >> 4;
    const int mt = blockIdx.x * 4 + wave;   // 16-row tile index (4 waves/block)
    const int nt = blockIdx.y;              // 16-col tile index
    if (mt >= Mtiles) return;               // wave-uniform exit

    const int m0  = mt << 4;
    const int n0  = nt << 4;
    const int row = m0 + lm;                // A row this lane feeds
    const int col = n0 + lm;                // B/C column this lane feeds

    v8f c = {};
    const float* arow = A + (size_t)row * K;
    for (int k = 0; k < K; k += 4) {
        // A fragment: contiguous pair (row, k+2*hi .. +1)
        v2f a = *(const v2f*)(arow + k + 2 * hi);
        // B fragment: strided column reads (tiny B stays hot in WGP$/L2)
        v2f b;
        b[0] = B[(size_t)(k + 2 * hi)     * N + col];
        b[1] = B[(size_t)(k + 2 * hi + 1) * N + col];
        c = __builtin_amdgcn_wmma_f32_16x16x4_f32(false, a, false, b,
                                                  (short)0, c, false, false);
    }
    #pragma unroll
    for (int v = 0; v < 8; ++v)
        C[(size_t)(m0 + v + 8 * hi) * N + col] = c[v];
}

// -------------------------------------------------------------------------
// SpMM segment-sum: Hout[r] = sum_{e in [row_ptr[r], row_ptr[r+1])}
//                             vals[e] * Hin[cols[e]]
// One wave per node row; each lane owns D/32 contiguous floats of the row
// (float4 for D=128, float2 for D=64) -> each gather is one full coalesced
// 512B/256B row read, served from L2 (all gather sources fit in 192MB L2).
// Edges are loaded cooperatively 32 at a time and broadcast via __shfl.
// Sequential accumulation => deterministic, no atomics.
// -------------------------------------------------------------------------
template <int D, bool RELU>
__global__ void gcn_spmm(const int*   __restrict__ row_ptr,
                         const int*   __restrict__ cols,
                         const float* __restrict__ vals,
                         const float* __restrict__ Hin,
                         float*       __restrict__ Hout) {
    const int lane = threadIdx.x & 31;
    const int row  = blockIdx.x * (blockDim.x >> 5) + (threadIdx.x >> 5);
    if (row >= GCN_N_NODES) return;

    const int beg = row_ptr[row];
    const int end = row_ptr[row + 1];

    constexpr int VPL = D / 32;  // floats per lane (4 or 2)
    float acc[VPL];
    #pragma unroll
    for (int i = 0; i < VPL; ++i) acc[i] = 0.0f;

    for (int base = beg; base < end; base += 32) {
        const int e   = base + lane;
        const int cnt = min(32, end - base);
        int   c_l = 0;
        float v_l = 0.0f;
        if (e < end) { c_l = cols[e]; v_l = vals[e]; }
        if (base + 32 < end) {  // prefetch next edge chunk (global_prefetch_b8)
            __builtin_prefetch(&cols[base + 32 + lane], 0, 0);
            __builtin_prefetch(&vals[base + 32 + lane], 0, 0);
        }
        for (int j = 0; j < cnt; ++j) {
            const int   cj = __shfl(c_l, j, 32);
            const float vj = __shfl(v_l, j, 32);
            const float* src = Hin + (size_t)cj * D + lane * VPL;
            if constexpr (VPL == 4) {
                float4 g = *(const float4*)src;
                acc[0] = fmaf(vj, g.x, acc[0]);
                acc[1] = fmaf(vj, g.y, acc[1]);
                acc[2] = fmaf(vj, g.z, acc[2]);
                acc[3] = fmaf(vj, g.w, acc[3]);
            } else {
                float2 g = *(const float2*)src;
                acc[0] = fmaf(vj, g.x, acc[0]);
                acc[1] = fmaf(vj, g.y, acc[1]);
            }
        }
    }

    float* dst = Hout + (size_t)row * D + lane * VPL;
    #pragma unroll
    for (int i = 0; i < VPL; ++i) {
        float o = acc[i];
        if (RELU) o = (o > 0.0f) ? o : 0.0f;
        dst[i] = o;
    }
}

// -------------------------------------------------------------------------
// Pipeline: row_ptr -> GEMM1(WMMA) -> SpMM+ReLU -> GEMM2(WMMA) -> SpMM -> out
// Workspace: XW1[100000x128] | H[100000x128] | row_ptr[100001]; HW2 aliases
// XW1 once it is dead. Peak ws use ~103 MB; intermediates stay L2-resident.
// -------------------------------------------------------------------------
extern "C" void kernel_launch(void* const* d_in, const int* in_sizes, int n_in,
                              void* d_out, int out_size, void* d_ws, size_t ws_size,
                              hipStream_t stream) {
    const float* x  = (const float*)d_in[0];  // [100000,256]
    const float* W1 = (const float*)d_in[1];  // [256,128]
    const float* W2 = (const float*)d_in[2];  // [128,64]
    const float* ev = (const float*)d_in[3];  // [E]
    const int*   er = (const int*)  d_in[4];  // [E] sorted
    const int*   ec = (const int*)  d_in[5];  // [E]
    float*       out = (float*)d_out;         // [100000,64]

    float* xw1  = (float*)d_ws;                                   // [N,128]
    float* h    = xw1 + (size_t)GCN_N_NODES * GCN_EMB;            // [N,128]
    int*   rptr = (int*)(h + (size_t)GCN_N_NODES * GCN_EMB);      // [N+1]
    float* hw2  = xw1;                                            // [N,64] reuse

    gcn_build_row_ptr<<<(GCN_N_EDGES + 255) / 256, 256, 0, stream>>>(
        er, rptr, GCN_N_EDGES, GCN_N_NODES);

    {   // XW1 = X @ W1   (M=100000, N=128, K=256)
        dim3 grid((GCN_N_NODES / 16 + 3) / 4, GCN_EMB / 16);
        gcn_gemm_wmma_f32<<<grid, 128, 0, stream>>>(
            x, W1, xw1, GCN_N_NODES, GCN_EMB, GCN_FEAT);
    }
    // H = relu(A @ XW1)
    gcn_spmm<GCN_EMB, true><<<GCN_N_NODES / 4, 128, 0, stream>>>(
        rptr, ec, ev, xw1, h);
    {   // HW2 = H @ W2   (M=100000, N=64, K=128)
        dim3 grid((GCN_N_NODES / 16 + 3) / 4, GCN_NCLS / 16);
        gcn_gemm_wmma_f32<<<grid, 128, 0, stream>>>(
            h, W2, hw2, GCN_N_NODES, GCN_NCLS, GCN_EMB);
    }
    // out = A @ HW2
    gcn_spmm<GCN_NCLS, false><<<GCN_N_NODES / 4, 128, 0, stream>>>(
        rptr, ec, ev, hw2, out);
}